// DGCNN_80169859547558
// MI455X (gfx1250) — compile-verified
//
#include <hip/hip_runtime.h>
#include <hip/hip_bf16.h>

// ---------------------------------------------------------------------------
// DGCNN forward for MI455X (gfx1250, wave32). FP32 end-to-end using
// V_WMMA_F32_16X16X4_F32 for all GEMM layers (memory-bound at fp32, so the
// fp32 matrix pipe is sufficient and preserves reference numerics).
// ---------------------------------------------------------------------------

typedef float v2f __attribute__((ext_vector_type(2)));
typedef float v8f __attribute__((ext_vector_type(8)));

#define NPTS   4096
#define KNN_K  25
#define NKCOLS (NPTS * KNN_K)

// ---------------- helpers ----------------

__device__ __forceinline__ float mishf(float v) {
    float sp = log1pf(expf(v));     // softplus; v>88 -> inf -> tanh=1 -> v (correct limit)
    return v * tanhf(sp);
}

// order-preserving float<->uint encode for atomic max over signed floats
__device__ __forceinline__ unsigned encOrd(float f) {
    unsigned u = __float_as_uint(f);
    return u ^ (unsigned)(((int)u >> 31) | 0x80000000);
}
__device__ __forceinline__ float decOrd(unsigned e) {
    unsigned m = (e & 0x80000000u) ? 0x80000000u : 0xFFFFFFFFu;
    return __uint_as_float(e ^ m);
}

// ---------------- GEMM: Y(OxM) = W(OxC, ld=ldw) * X(CxM), fp32 WMMA ----------
// REQUIREMENT: M % 128 == 0 (true for all call sites: 4096, 102400).
// mode 0: plain store Y
// mode 1: Y = mish((dot [+ accBias[row]]) * scale[row] + bias[row])
// mode 2: val as mode 1, then atomic-max-encode into Yenc[row, col/maxDiv]
#define TILE_M  128
#define TILE_O  128
#define TILE_C  32
#define PAIR_LD 144  // float2 units; 2*144 % 64 == 32 -> lane halves conflict-free

__global__ __launch_bounds__(256) void gemm_wmma_kernel(
    const float* __restrict__ W, int ldw,
    const float* __restrict__ X,
    const float* __restrict__ scale, const float* __restrict__ bias,
    const float* __restrict__ accBias,
    float* __restrict__ Y, unsigned* __restrict__ Yenc,
    int O, int C, int M, int mode, int maxDiv)
{
    // row-pair interleaved X tile: Xp[p][col] = (row 2p, row 2p+1) at col
    __shared__ __align__(16) float2 Xp[TILE_C / 2][PAIR_LD];

    const int tid  = threadIdx.x;
    const int lane = tid & 31;
    const int wave = tid >> 5;
    const int l16  = lane & 15;
    const int khalf = (lane >> 4) << 1;           // 0 (lanes 0-15) or 2 (lanes 16-31)
    const int colBase = blockIdx.x * TILE_M;
    const int rowBase = blockIdx.y * TILE_O + wave * 16;
    const int arow = rowBase + l16;               // A-fragment row for this lane
    const int arowC = (arow < O) ? arow : (O - 1);
    const bool rowOk = (arow < O);
    const float* __restrict__ wrow = W + (size_t)arowC * ldw;

    v8f acc[8];
    v8f z = {};
#pragma unroll
    for (int j = 0; j < 8; ++j) acc[j] = z;

    // cooperative-load geometry: each thread loads float4 at 4 rows
    const int colQ = (tid & 31) * 4;
    const int rowL = tid >> 5;
    const float* __restrict__ Xcol = X + colBase + colQ;

    auto loadChunk = [&](int cB, float4* r) {
        if (cB + TILE_C <= C) {                               // fast path (uniform)
#pragma unroll
            for (int i = 0; i < 4; ++i)
                r[i] = *(const float4*)(Xcol + (size_t)(cB + rowL + i * 8) * M);
        } else {                                              // zero-padded tail
#pragma unroll
            for (int i = 0; i < 4; ++i) {
                int gc = cB + rowL + i * 8;
                int gcc = (gc < C) ? gc : (C - 1);
                float msk = (gc < C) ? 1.0f : 0.0f;
                float4 vv = *(const float4*)(Xcol + (size_t)gcc * M);
                r[i] = make_float4(vv.x * msk, vv.y * msk, vv.z * msk, vv.w * msk);
            }
        }
    };

    float4 stage[4];
    loadChunk(0, stage);

    auto storeStage = [&]() {
#pragma unroll
        for (int i = 0; i < 4; ++i) {
            int cL = rowL + i * 8;
            float* bp = ((float*)Xp) + (size_t)(cL >> 1) * (PAIR_LD * 2) + (cL & 1);
            bp[(colQ + 0) * 2] = stage[i].x;
            bp[(colQ + 1) * 2] = stage[i].y;
            bp[(colQ + 2) * 2] = stage[i].z;
            bp[(colQ + 3) * 2] = stage[i].w;
        }
    };

    int cBase = 0;
    for (; cBase + TILE_C <= C; cBase += TILE_C) {     // aligned chunks: single path
        __syncthreads();
        storeStage();
        __syncthreads();

        int cNext = cBase + TILE_C;
        if (cNext < C) loadChunk(cNext, stage);        // prefetch overlaps the WMMAs

#pragma unroll
        for (int c4 = 0; c4 < TILE_C; c4 += 4) {
            int gc0 = cBase + c4 + khalf;
            v2f a;
            a.x = rowOk ? wrow[gc0]     : 0.0f;
            a.y = rowOk ? wrow[gc0 + 1] : 0.0f;
            const int p = (c4 + khalf) >> 1;
#pragma unroll
            for (int j = 0; j < 8; ++j) {
                v2f b = *(const v2f*)&Xp[p][j * 16 + l16];
                acc[j] = __builtin_amdgcn_wmma_f32_16x16x4_f32(
                    false, a, false, b, (short)0, acc[j], false, false);
            }
        }
    }

    if (cBase < C) {                                   // ragged tail chunk (<= 1)
        __syncthreads();
        storeStage();
        __syncthreads();
#pragma unroll
        for (int c4 = 0; c4 < TILE_C; c4 += 4) {
            int gc0 = cBase + c4 + khalf;
            int cx = (gc0     < C) ? gc0     : (C - 1);
            int cy = (gc0 + 1 < C) ? gc0 + 1 : (C - 1);
            float ax = wrow[cx];
            float ay = wrow[cy];
            v2f a;                                     // zeroing required: 0*Inf -> NaN
            a.x = (rowOk && gc0     < C) ? ax : 0.0f;
            a.y = (rowOk && gc0 + 1 < C) ? ay : 0.0f;
            const int p = (c4 + khalf) >> 1;
#pragma unroll
            for (int j = 0; j < 8; ++j) {
                v2f b = *(const v2f*)&Xp[p][j * 16 + l16];
                acc[j] = __builtin_amdgcn_wmma_f32_16x16x4_f32(
                    false, a, false, b, (short)0, acc[j], false, false);
            }
        }
    }

    // epilogue: lane holds (per acc[j]) rows rowBase + jj + (lane>=16 ? 8:0)
    const int rowOff = (lane >> 4) * 8;

    if (mode == 2 && maxDiv >= TILE_M) {
        // whole block maps to a single reduce-group per row: pre-reduce in regs
        int g0 = colBase / maxDiv;
        size_t gstride = (size_t)(M / maxDiv);
#pragma unroll
        for (int jj = 0; jj < 8; ++jj) {
            int r = rowBase + rowOff + jj;
            if (r >= O) break;
            float ab = accBias ? accBias[r] : 0.0f;
            float s = scale[r], bb = bias[r];
            float m = -3.4e38f;
#pragma unroll
            for (int j = 0; j < 8; ++j)
                m = fmaxf(m, mishf((acc[j][jj] + ab) * s + bb));
            atomicMax(&Yenc[(size_t)r * gstride + g0], encOrd(m));
        }
        return;
    }

#pragma unroll
    for (int j = 0; j < 8; ++j) {
        int col = colBase + j * 16 + l16;
#pragma unroll
        for (int jj = 0; jj < 8; ++jj) {
            int r = rowBase + rowOff + jj;
            if (r >= O) break;
            float val = acc[j][jj];
            if (mode == 0) {
                Y[(size_t)r * M + col] = val;
            } else {
                if (accBias) val += accBias[r];
                val = mishf(val * scale[r] + bias[r]);
                if (mode == 1) {
                    Y[(size_t)r * M + col] = val;
                } else {
                    int g = col / maxDiv;
                    atomicMax(&Yenc[(size_t)r * (size_t)(M / maxDiv) + g], encOrd(val));
                }
            }
        }
    }
}

// ---------------- squared norms ----------------
__global__ void norms_kernel(const float* __restrict__ x, float* __restrict__ xx,
                             int C, int N)
{
    int n = blockIdx.x * blockDim.x + threadIdx.x;
    if (n >= N) return;
    float s = 0.f;
    for (int c = 0; c < C; ++c) { float v = x[(size_t)c * N + n]; s += v * v; }
    xx[n] = s;
}

// ---------------- kNN: per-thread top-25 of pd = 2*dot - ||a||^2 - ||b||^2 ----
template <int C>
__global__ __launch_bounds__(128) void knn_kernel(
    const float* __restrict__ X, const float* __restrict__ xx,
    int* __restrict__ idx, int N)
{
    constexpr int TILE = 128;
    __shared__ float Xt[C][TILE];
    __shared__ float xxs[TILE];
    __shared__ float tv[128][KNN_K + 1];
    __shared__ int   ti[128][KNN_K + 1];

    const int t = threadIdx.x;
    const int n = blockIdx.x * 128 + t;

    float r[C];
#pragma unroll
    for (int c = 0; c < C; ++c) r[c] = X[(size_t)c * N + n];
    const float xn = xx[n];

    for (int j = 0; j < KNN_K; ++j) { tv[t][j] = -3.4e38f; ti[t][j] = 0; }

    for (int mBase = 0; mBase < N; mBase += TILE) {
        __syncthreads();
        for (int e = t; e < C * TILE; e += 128) {
            int cc = e / TILE, mm = e % TILE;
            Xt[cc][mm] = X[(size_t)cc * N + mBase + mm];
        }
        xxs[t] = xx[mBase + t];
        __syncthreads();
        for (int mm = 0; mm < TILE; ++mm) {
            float dot = 0.f;
#pragma unroll
            for (int c = 0; c < C; ++c) dot += r[c] * Xt[c][mm];
            float d = 2.f * dot - xn - xxs[mm];
            if (d > tv[t][KNN_K - 1]) {
                int pos = KNN_K - 1;
                while (pos > 0 && tv[t][pos - 1] < d) {   // strict: low-index tie-break
                    tv[t][pos] = tv[t][pos - 1]; ti[t][pos] = ti[t][pos - 1]; --pos;
                }
                tv[t][pos] = d; ti[t][pos] = mBase + mm;
            }
        }
    }
    for (int j = 0; j < KNN_K; ++j) idx[(size_t)n * KNN_K + j] = ti[t][j];
}

// ---------------- edge gather: h[o,nk] = mish((u[o,idx]+v[o,n])*s+b) ----------
// grid: (NK/256, O)
__global__ void gather_edge_kernel(
    const float* __restrict__ u, const float* __restrict__ v,
    const int* __restrict__ idx,
    const float* __restrict__ scale, const float* __restrict__ bias,
    float* __restrict__ h, int N, int NK)
{
    int nk = blockIdx.x * blockDim.x + threadIdx.x;
    if (nk >= NK) return;
    int o = blockIdx.y;
    int n = nk / KNN_K;
    int j = idx[nk];
    float y = u[(size_t)o * N + j] + v[(size_t)o * N + n];
    h[(size_t)o * NK + nk] = mishf(y * scale[o] + bias[o]);
}

// ---------------- w_diff = w[:, Ci:2Ci] - w[:, 0:Ci] --------------------------
__global__ void wdiff_kernel(const float* __restrict__ w, float* __restrict__ wd,
                             int O, int Ci)
{
    int t = blockIdx.x * blockDim.x + threadIdx.x;
    if (t >= O * Ci) return;
    int o = t / Ci, c = t % Ci;
    wd[t] = w[(size_t)o * 2 * Ci + Ci + c] - w[(size_t)o * 2 * Ci + c];
}

// ---------------- decode atomic-max buffers ----------------------------------
__global__ void decode_kernel(const unsigned* __restrict__ e, float* __restrict__ out, int n)
{
    int i = blockIdx.x * blockDim.x + threadIdx.x;
    if (i < n) out[i] = decOrd(e[i]);
}

// ---------------- small dense: y[o] = act(sum_c W[o,c]*x[c] + b[o]) -----------
__global__ void fc_kernel(const float* __restrict__ W, int ldw,
                          const float* __restrict__ x, const float* __restrict__ bias,
                          float* __restrict__ y, int O, int C, int act)
{
    int o = blockIdx.x * blockDim.x + threadIdx.x;
    if (o >= O) return;
    float s = bias ? bias[o] : 0.f;
    for (int c = 0; c < C; ++c) s += W[(size_t)o * ldw + c] * x[c];
    y[o] = act ? mishf(s) : s;
}

// ---------------- apply 15x15 transform: y[e,n] = sum_c x[c,n]*t[c*15+e] ------
__global__ void tfapply_kernel(const float* __restrict__ x, const float* __restrict__ t,
                               float* __restrict__ y, int N)
{
    int n = blockIdx.x * blockDim.x + threadIdx.x;
    if (n >= N) return;
    float xv[15];
#pragma unroll
    for (int c = 0; c < 15; ++c) xv[c] = x[(size_t)c * N + n];
#pragma unroll
    for (int e = 0; e < 15; ++e) {
        float s = 0.f;
#pragma unroll
        for (int c = 0; c < 15; ++c) s += xv[c] * t[c * 15 + e];
        y[(size_t)e * N + n] = s;
    }
}

// ---------------------------------------------------------------------------

static inline void gemm(const float* W, int ldw, const float* X,
                        const float* s, const float* b, const float* ab,
                        float* Y, unsigned* Yenc,
                        int O, int C, int M, int mode, int maxDiv, hipStream_t st)
{
    dim3 grid((M + TILE_M - 1) / TILE_M, (O + TILE_O - 1) / TILE_O);
    gemm_wmma_kernel<<<grid, 256, 0, st>>>(W, ldw, X, s, b, ab, Y, Yenc, O, C, M, mode, maxDiv);
}

extern "C" void kernel_launch(void* const* d_in, const int* in_sizes, int n_in,
                              void* d_out, int out_size, void* d_ws, size_t ws_size,
                              hipStream_t stream)
{
    const int N = NPTS, NK = NKCOLS, B = 4;
    (void)in_sizes; (void)n_in; (void)out_size; (void)ws_size;

    int ii = 0;
    const float* Xin      = (const float*)d_in[ii++];          // (4,15,4096)
    const float* tn_c1_w  = (const float*)d_in[ii++];
    const float* tn_c1_s  = (const float*)d_in[ii++];
    const float* tn_c1_b  = (const float*)d_in[ii++];
    const float* tn_c2_w  = (const float*)d_in[ii++];
    const float* tn_c2_s  = (const float*)d_in[ii++];
    const float* tn_c2_b  = (const float*)d_in[ii++];
    const float* tn_c3_w  = (const float*)d_in[ii++];
    const float* tn_c3_s  = (const float*)d_in[ii++];
    const float* tn_c3_b  = (const float*)d_in[ii++];
    const float* tn_l1_w  = (const float*)d_in[ii++];
    const float* tn_l1_b  = (const float*)d_in[ii++];
    const float* tn_l2_w  = (const float*)d_in[ii++];
    const float* tn_l2_b  = (const float*)d_in[ii++];
    const float* tn_tf_w  = (const float*)d_in[ii++];
    const float* tn_tf_b  = (const float*)d_in[ii++];
    const float* c1_w = (const float*)d_in[ii++]; const float* c1_s = (const float*)d_in[ii++]; const float* c1_b = (const float*)d_in[ii++];
    const float* c21_w = (const float*)d_in[ii++]; const float* c21_s = (const float*)d_in[ii++]; const float* c21_b = (const float*)d_in[ii++];
    const float* c22_w = (const float*)d_in[ii++]; const float* c22_s = (const float*)d_in[ii++]; const float* c22_b = (const float*)d_in[ii++];
    const float* c3_w = (const float*)d_in[ii++]; const float* c3_s = (const float*)d_in[ii++]; const float* c3_b = (const float*)d_in[ii++];
    const float* c41_w = (const float*)d_in[ii++]; const float* c41_s = (const float*)d_in[ii++]; const float* c41_b = (const float*)d_in[ii++];
    const float* c42_w = (const float*)d_in[ii++]; const float* c42_s = (const float*)d_in[ii++]; const float* c42_b = (const float*)d_in[ii++];
    const float* c51_w = (const float*)d_in[ii++]; const float* c51_s = (const float*)d_in[ii++]; const float* c51_b = (const float*)d_in[ii++];
    const float* c52_w = (const float*)d_in[ii++]; const float* c52_s = (const float*)d_in[ii++]; const float* c52_b = (const float*)d_in[ii++];
    const float* c6_w = (const float*)d_in[ii++]; const float* c6_s = (const float*)d_in[ii++]; const float* c6_b = (const float*)d_in[ii++];
    const float* p1_w = (const float*)d_in[ii++]; const float* p1_s = (const float*)d_in[ii++]; const float* p1_b = (const float*)d_in[ii++];
    const float* p2_w = (const float*)d_in[ii++];
    float* out = (float*)d_out;

    // ---- workspace bump allocator ----
    char* wsp = (char*)d_ws;
    auto alloc = [&](size_t bytes) -> void* {
        void* p = (void*)wsp;
        wsp += (bytes + 255) & ~(size_t)255;
        return p;
    };
    float* wd_tn  = (float*)alloc(128 * 15 * 4);
    float* wd_c1  = (float*)alloc(128 * 15 * 4);
    float* wd_c3  = (float*)alloc(64 * 64 * 4);
    float* wd_c51 = (float*)alloc(64 * 64 * 4);
    float* norms  = (float*)alloc((size_t)N * 4);
    int*   idx    = (int*)  alloc((size_t)NK * 4);
    float* u      = (float*)alloc((size_t)128 * N * 4);
    float* v      = (float*)alloc((size_t)128 * N * 4);
    float* h1     = (float*)alloc((size_t)128 * NK * 4);   // 52.4 MB
    float* h2     = (float*)alloc((size_t)128 * NK * 4);   // 52.4 MB
    unsigned* enc = (unsigned*)alloc((size_t)256 * N * 4); // 4 MB
    float* t2m    = (float*)alloc((size_t)256 * N * 4);
    float* vec1k  = (float*)alloc(1024 * 4);
    float* l1o    = (float*)alloc(512 * 4);
    float* l2o    = (float*)alloc(256 * 4);
    float* tfo    = (float*)alloc(256 * 4);
    float* pg     = (float*)alloc(512 * 4);
    float* xt     = (float*)alloc((size_t)15 * N * 4);
    float* cat    = (float*)alloc((size_t)192 * N * 4);    // x1|x2|x3
    float* hp1    = (float*)alloc((size_t)512 * N * 4);

    float* x1 = cat;
    float* x2 = cat + (size_t)64 * N;
    float* x3 = cat + (size_t)128 * N;

    // factored first-conv weights (batch independent)
    wdiff_kernel<<<(128 * 15 + 255) / 256, 256, 0, stream>>>(tn_c1_w, wd_tn, 128, 15);
    wdiff_kernel<<<(128 * 15 + 255) / 256, 256, 0, stream>>>(c1_w,    wd_c1, 128, 15);
    wdiff_kernel<<<(64 * 64 + 255) / 256, 256, 0, stream>>>(c3_w,  wd_c3,  64, 64);
    wdiff_kernel<<<(64 * 64 + 255) / 256, 256, 0, stream>>>(c51_w, wd_c51, 64, 64);

    for (int b = 0; b < B; ++b) {
        const float* xb = Xin + (size_t)b * 15 * N;

        // ---- knn on xyz ----
        norms_kernel<<<(N + 255) / 256, 256, 0, stream>>>(xb, norms, 3, N);
        knn_kernel<3><<<N / 128, 128, 0, stream>>>(xb, norms, idx, N);

        // ---- transform net ----
        gemm(tn_c1_w, 30, xb, 0, 0, 0, u, 0, 128, 15, N, 0, 1, stream);
        gemm(wd_tn,   15, xb, 0, 0, 0, v, 0, 128, 15, N, 0, 1, stream);
        gather_edge_kernel<<<dim3(NK / 256, 128), 256, 0, stream>>>(
            u, v, idx, tn_c1_s, tn_c1_b, h1, N, NK);
        hipMemsetAsync(enc, 0, (size_t)256 * N * 4, stream);
        gemm(tn_c2_w, 128, h1, tn_c2_s, tn_c2_b, 0, 0, enc, 256, 128, NK, 2, KNN_K, stream);
        decode_kernel<<<(256 * N + 255) / 256, 256, 0, stream>>>(enc, t2m, 256 * N);
        hipMemsetAsync(enc, 0, 1024 * 4, stream);
        gemm(tn_c3_w, 256, t2m, tn_c3_s, tn_c3_b, 0, 0, enc, 1024, 256, N, 2, N, stream);
        decode_kernel<<<4, 256, 0, stream>>>(enc, vec1k, 1024);
        fc_kernel<<<2, 256, 0, stream>>>(tn_l1_w, 1024, vec1k, tn_l1_b, l1o, 512, 1024, 1);
        fc_kernel<<<1, 256, 0, stream>>>(tn_l2_w, 512,  l1o,   tn_l2_b, l2o, 256, 512, 1);
        fc_kernel<<<1, 256, 0, stream>>>(tn_tf_w, 256,  l2o,   tn_tf_b, tfo, 225, 256, 0);
        tfapply_kernel<<<N / 256, 256, 0, stream>>>(xb, tfo, xt, N);

        // ---- EdgeConv 1 (xt, 15ch) ----
        norms_kernel<<<(N + 255) / 256, 256, 0, stream>>>(xt, norms, 15, N);
        knn_kernel<15><<<N / 128, 128, 0, stream>>>(xt, norms, idx, N);
        gemm(c1_w,  30, xt, 0, 0, 0, u, 0, 128, 15, N, 0, 1, stream);
        gemm(wd_c1, 15, xt, 0, 0, 0, v, 0, 128, 15, N, 0, 1, stream);
        gather_edge_kernel<<<dim3(NK / 256, 128), 256, 0, stream>>>(
            u, v, idx, c1_s, c1_b, h1, N, NK);
        gemm(c21_w, 128, h1, c21_s, c21_b, 0, h2, 0, 128, 128, NK, 1, 1, stream);
        hipMemsetAsync(enc, 0, (size_t)64 * N * 4, stream);
        gemm(c22_w, 128, h2, c22_s, c22_b, 0, 0, enc, 64, 128, NK, 2, KNN_K, stream);
        decode_kernel<<<(64 * N + 255) / 256, 256, 0, stream>>>(enc, x1, 64 * N);

        // ---- EdgeConv 2 (x1, 64ch) ----
        norms_kernel<<<(N + 255) / 256, 256, 0, stream>>>(x1, norms, 64, N);
        knn_kernel<64><<<N / 128, 128, 0, stream>>>(x1, norms, idx, N);
        gemm(c3_w,  128, x1, 0, 0, 0, u, 0, 64, 64, N, 0, 1, stream);
        gemm(wd_c3,  64, x1, 0, 0, 0, v, 0, 64, 64, N, 0, 1, stream);
        gather_edge_kernel<<<dim3(NK / 256, 64), 256, 0, stream>>>(
            u, v, idx, c3_s, c3_b, h1, N, NK);
        gemm(c41_w, 64, h1, c41_s, c41_b, 0, h2, 0, 64, 64, NK, 1, 1, stream);
        hipMemsetAsync(enc, 0, (size_t)64 * N * 4, stream);
        gemm(c42_w, 64, h2, c42_s, c42_b, 0, 0, enc, 64, 64, NK, 2, KNN_K, stream);
        decode_kernel<<<(64 * N + 255) / 256, 256, 0, stream>>>(enc, x2, 64 * N);

        // ---- EdgeConv 3 (x2, 64ch) ----
        norms_kernel<<<(N + 255) / 256, 256, 0, stream>>>(x2, norms, 64, N);
        knn_kernel<64><<<N / 128, 128, 0, stream>>>(x2, norms, idx, N);
        gemm(c51_w, 128, x2, 0, 0, 0, u, 0, 64, 64, N, 0, 1, stream);
        gemm(wd_c51, 64, x2, 0, 0, 0, v, 0, 64, 64, N, 0, 1, stream);
        gather_edge_kernel<<<dim3(NK / 256, 64), 256, 0, stream>>>(
            u, v, idx, c51_s, c51_b, h1, N, NK);
        hipMemsetAsync(enc, 0, (size_t)64 * N * 4, stream);
        gemm(c52_w, 64, h1, c52_s, c52_b, 0, 0, enc, 64, 64, NK, 2, KNN_K, stream);
        decode_kernel<<<(64 * N + 255) / 256, 256, 0, stream>>>(enc, x3, 64 * N);

        // ---- global feature + head ----
        hipMemsetAsync(enc, 0, 1024 * 4, stream);
        gemm(c6_w, 192, cat, c6_s, c6_b, 0, 0, enc, 1024, 192, N, 2, N, stream);
        decode_kernel<<<4, 256, 0, stream>>>(enc, vec1k, 1024);
        // pg[o] = p1_w[:, :1024] @ gmax   (constant over n, folded as acc-bias)
        fc_kernel<<<2, 256, 0, stream>>>(p1_w, 1216, vec1k, nullptr, pg, 512, 1024, 0);
        gemm(p1_w + 1024, 1216, cat, p1_s, p1_b, pg, hp1, 0, 512, 192, N, 1, 1, stream);
        // final projection straight into d_out
        gemm(p2_w, 512, hp1, 0, 0, 0, out + (size_t)b * 32 * N, 0, 32, 512, N, 0, 1, stream);
    }
}